// STFT_5669356831421
// MI455X (gfx1250) — compile-verified
//
#include <hip/hip_runtime.h>

// ---------------- problem constants ----------------
#define FL    800
#define HOP   200
#define CUT   401
#define NCH   802            // 2*CUT channels
#define NB    16
#define T_IN  160000
#define NF    801            // frames
#define NN    (NB * NF)      // 12816 GEMM columns (multiple of 16)
#define MPAD1 816            // 802 padded to 16

typedef float v2f __attribute__((ext_vector_type(2)));
typedef float v8f __attribute__((ext_vector_type(8)));

// A tile: 128 x 16 stored [m][k], even pitch -> lane fragment pair is one b64
#define APITCH 18
// B tile: 16 x 128 stored fragment-ready: [kchunk][ntile][lane][pair]

// MODE 0: C(816x12816) = fwd(802x800, zero-padded M) * patches(800x12816)
//          patches built on the fly from reflect-padded input
// MODE 1: C(800x12816) = inv^T(800x816) * ft(816x12816)   (A read transposed)
template <int MODE>
__global__ __launch_bounds__(256)
void stft_gemm(const float* __restrict__ Abase,
               const float* __restrict__ Bbase,
               float* __restrict__ Cout,
               int Mvalid, int Mstore, int Ksz)
{
    __shared__ float As[128 * APITCH];     // 9216 B
    __shared__ float Bs[4 * 8 * 32 * 2];   // 8192 B (2048 floats)

    const int tid   = threadIdx.x;
    const int lane  = tid & 31;
    const int wave  = tid >> 5;
    const int mBase = blockIdx.y * 128;
    const int nBase = blockIdx.x * 128;

    v8f acc[8] = {};

    for (int kb = 0; kb < Ksz; kb += 16) {
        // ---- stage A tile (128 x 16) into LDS [m][k], zero-fill OOR ----
        #pragma unroll
        for (int i = 0; i < 8; ++i) {
            int idx = tid + i * 256;           // 0..2047
            int k, m;
            float v = 0.0f;
            if (MODE == 0) {
                k = idx & 15; m = idx >> 4;    // k fast -> coalesced (row-major A)
                int gm = mBase + m;
                if (gm < Mvalid) v = Abase[gm * FL + kb + k];   // kb+k <= 799
            } else {
                m = idx & 127; k = idx >> 7;   // m fast -> coalesced (transposed A)
                int gm = mBase + m, gk = kb + k;
                if (gm < Mvalid && gk < NCH) v = Abase[gk * FL + gm];
            }
            As[m * APITCH + k] = v;
        }
        // ---- stage B tile (16 x 128) into fragment-ready LDS layout ----
        // lane L of chunk kc, tile nt holds pair (K=4kc+(L>>4), K=4kc+2+(L>>4))
        #pragma unroll
        for (int i = 0; i < 8; ++i) {
            int idx = tid + i * 256;           // 0..2047
            int k, nn;
            float v = 0.0f;
            if (MODE == 0) {
                k = idx & 15; nn = idx >> 4;   // k fast: contiguous within a frame
                int n = nBase + nn;
                if (n < NN) {
                    int b = n / NF, f = n - b * NF;
                    int pos = f * HOP + kb + k - FL / 2;        // reflect pad
                    pos = (pos < 0) ? -pos : pos;
                    pos = (pos >= T_IN) ? (2 * (T_IN - 1) - pos) : pos;
                    v = Bbase[b * T_IN + pos];
                }
            } else {
                nn = idx & 127; k = idx >> 7;  // n fast -> coalesced rows of ft
                int n = nBase + nn;
                if (n < NN) v = Bbase[(kb + k) * NN + n];       // kb+k < 816
            }
            int chunk = k >> 2, kr = k & 3;
            int hi = kr & 1, sel = kr >> 1;
            int nt = nn >> 4, c = nn & 15;
            Bs[(((chunk << 3) + nt) * 32 + (hi << 4) + c) * 2 + sel] = v;
        }
        __syncthreads();

        // ---- 4 k-chunks x 8 n-tiles of V_WMMA_F32_16X16X4_F32 ----
        const int mloc = wave * 16 + (lane & 15);
        const int hi   = lane >> 4;            // 0 or 1 (half-wave)
        const v2f* Bf  = (const v2f*)Bs;
        #pragma unroll
        for (int kc = 0; kc < 4; ++kc) {
            // A 16x4 frag: lanes0-15 K=k0,k0+1; lanes16-31 K=k0+2,k0+3
            v2f a = *(const v2f*)&As[mloc * APITCH + kc * 4 + 2 * hi];
            #pragma unroll
            for (int nt = 0; nt < 8; ++nt) {
                v2f bf = Bf[((kc << 3) + nt) * 32 + lane];
                acc[nt] = __builtin_amdgcn_wmma_f32_16x16x4_f32(
                    false, a, false, bf, (short)0, acc[nt], false, false);
            }
        }
        __syncthreads();
    }

    // ---- store: C/D layout: VGPR j -> M = j + 8*(lane>>4), N = lane&15 ----
    const int m0 = mBase + wave * 16 + (lane >> 4) * 8;
    const int c0 = lane & 15;
    #pragma unroll
    for (int nt = 0; nt < 8; ++nt) {
        int col = nBase + nt * 16 + c0;
        if (col < NN) {
            #pragma unroll
            for (int j = 0; j < 8; ++j) {
                int row = m0 + j;
                if (row < Mstore) Cout[row * NN + col] = acc[nt][j];
            }
        }
    }
}

// ---- overlap-add + window-sum-square normalization + crop ----
__global__ __launch_bounds__(256)
void overlap_add(const float* __restrict__ Z, float* __restrict__ out)
{
    int idx = blockIdx.x * blockDim.x + threadIdx.x;
    if (idx >= NB * T_IN) return;
    int b = idx / T_IN, t = idx - b * T_IN;
    int tp = t + FL / 2;                               // position in padded signal
    int fmax = tp / HOP;           if (fmax > NF - 1) fmax = NF - 1;
    int fmin = (tp >= FL) ? ((tp - (FL - 1) + HOP - 1) / HOP) : 0;

    float acc = 0.0f, wsum = 0.0f;
    #pragma unroll 4
    for (int f = fmin; f <= fmax; ++f) {
        int k = tp - f * HOP;                          // 0..FL-1
        acc += Z[k * NN + b * NF + f];
        float w = 0.5f - 0.5f * cosf(6.2831853071795864f * (float)k / (float)FL);
        wsum += w * w;
    }
    const float tiny = 1.1754944e-38f;
    float denom = (wsum > tiny) ? wsum : 1.0f;
    out[idx] = acc * ((float)FL / (float)HOP) / denom;
}

extern "C" void kernel_launch(void* const* d_in, const int* in_sizes, int n_in,
                              void* d_out, int out_size, void* d_ws, size_t ws_size,
                              hipStream_t stream)
{
    const float* x   = (const float*)d_in[0];   // 16 x 160000
    const float* fwd = (const float*)d_in[1];   // 802 x 800
    const float* inv = (const float*)d_in[2];   // 802 x 800
    float* out = (float*)d_out;                 // 16 x 1 x 160000

    float* ft = (float*)d_ws;                               // 816 x 12816
    float* Z  = ft + (size_t)MPAD1 * NN;                    // 800 x 12816

    dim3 blk(256);
    dim3 g1((NN + 127) / 128, (MPAD1 + 127) / 128);         // 101 x 7
    stft_gemm<0><<<g1, blk, 0, stream>>>(fwd, x, ft, NCH, MPAD1, FL);

    dim3 g2((NN + 127) / 128, (FL + 127) / 128);            // 101 x 7
    stft_gemm<1><<<g2, blk, 0, stream>>>(inv, ft, Z, FL, FL, MPAD1);

    int tot = NB * T_IN;
    overlap_add<<<(tot + 255) / 256, blk, 0, stream>>>(Z, out);
}